// GraphConvolution_27315992003075
// MI455X (gfx1250) — compile-verified
//
#include <hip/hip_runtime.h>

typedef __attribute__((ext_vector_type(2))) float v2f;
typedef __attribute__((ext_vector_type(8))) float v8f;

#define GCN_N_NODES 100000
#define GCN_IN_DIM  256
#define GCN_OUT_DIM 64
#define GCN_N_EDGES 1600000

// Padded row stride (in float2 units) for the K-pair-interleaved W tile.
// stride = 80 => word-bank offset between lane-half 0 (pair-row p) and
// lane-half 1 (pair-row p+1) is 2*80 mod 64 = 32 -> the ds_load_b64 hits all
// 64 banks conflict-free.
#define WPAD 80

// ---------------------------------------------------------------------------
// Kernel 1: prod = X @ W  using V_WMMA_F32_16X16X4_F32 (exact fp32 path).
// Block = 256 threads = 8 waves; each wave computes a 16-row x 64-col strip,
// so X is streamed exactly once. W is staged in LDS with K-pairs interleaved
// so each B operand is a single ds_load_b64 into an aligned VGPR pair.
// ---------------------------------------------------------------------------
__global__ __launch_bounds__(256) void gcn_gemm_xw(const float* __restrict__ X,
                                                   const float* __restrict__ W,
                                                   float* __restrict__ prod) {
    __shared__ float2 ldsP[(GCN_IN_DIM / 2) * WPAD];  // 128 pair-rows x 80 (80 KB)
    const int tid = threadIdx.x;

    // Stage W: ldsP[p][col] = {W[2p][col], W[2p+1][col]}
    for (int i = tid; i < (GCN_IN_DIM / 2) * GCN_OUT_DIM; i += 256) {
        const int p   = i >> 6;        // pair-row 0..127
        const int col = i & 63;
        ldsP[p * WPAD + col] =
            float2{W[(2 * p) * GCN_OUT_DIM + col], W[(2 * p + 1) * GCN_OUT_DIM + col]};
    }
    __syncthreads();

    const int wave = tid >> 5;
    const int lane = tid & 31;
    const int half = lane >> 4;   // 0: K={k,k+1}, rows v..v+7 ; 1: K={k+2,k+3}, rows +8
    const int mn   = lane & 15;

    const int rowBase = blockIdx.x * 128 + wave * 16;
    int arow = rowBase + mn;
    if (arow > GCN_N_NODES - 1) arow = GCN_N_NODES - 1;   // clamp: stores are masked
    const float* aptr = X + (size_t)arow * GCN_IN_DIM + 2 * half;

    v8f c0 = {}, c1 = {}, c2 = {}, c3 = {};

    for (int k = 0; k < GCN_IN_DIM; k += 4) {
        // A 16x4 tile: per-lane float2 at K = k + 2*half, +1
        v2f a = *(const v2f*)(aptr + k);

        // B 4x16 tiles: one ds_load_b64 per tile, pair-row = k/2 + half
        const float2* bp = &ldsP[((k >> 1) + half) * WPAD + mn];
        v2f b0 = *(const v2f*)(bp + 0);
        v2f b1 = *(const v2f*)(bp + 16);
        v2f b2 = *(const v2f*)(bp + 32);
        v2f b3 = *(const v2f*)(bp + 48);

        c0 = __builtin_amdgcn_wmma_f32_16x16x4_f32(false, a, false, b0, (short)0, c0, false, false);
        c1 = __builtin_amdgcn_wmma_f32_16x16x4_f32(false, a, false, b1, (short)0, c1, false, false);
        c2 = __builtin_amdgcn_wmma_f32_16x16x4_f32(false, a, false, b2, (short)0, c2, false, false);
        c3 = __builtin_amdgcn_wmma_f32_16x16x4_f32(false, a, false, b3, (short)0, c3, false, false);
    }

    // C/D layout: VGPR v -> row (v + 8*half), col = mn (+16 per tile)
#pragma unroll
    for (int v = 0; v < 8; ++v) {
        const int r = rowBase + v + 8 * half;
        if (r < GCN_N_NODES) {
            float* o = prod + (size_t)r * GCN_OUT_DIM + mn;
            o[0]  = c0[v];
            o[16] = c1[v];
            o[32] = c2[v];
            o[48] = c3[v];
        }
    }
}

// ---------------------------------------------------------------------------
// Kernel 2: zero the accumulator (d_out) — harness poisons it with 0xAA.
// ---------------------------------------------------------------------------
__global__ __launch_bounds__(256) void gcn_zero(float4* __restrict__ out) {
    const unsigned i = blockIdx.x * 256u + threadIdx.x;
    out[i] = float4{0.f, 0.f, 0.f, 0.f};
}

// ---------------------------------------------------------------------------
// Kernel 3: edge scatter: out[r] += val * prod[c].
// One edge per wave32 (edge metadata loads are wave-uniform -> one L2 request);
// each lane handles 2 columns: float2 gather + 2 hardware fp32 atomics in L2.
// ---------------------------------------------------------------------------
__global__ __launch_bounds__(256) void gcn_spmm(const int*   __restrict__ rows,
                                                const int*   __restrict__ cols,
                                                const float* __restrict__ vals,
                                                const float* __restrict__ prod,
                                                float*       __restrict__ out) {
    const unsigned gid  = blockIdx.x * 256u + threadIdx.x;
    const unsigned e    = gid >> 5;          // wave-uniform edge index
    if (e >= GCN_N_EDGES) return;
    const int lane = threadIdx.x & 31;

    const int   r = rows[e];
    const int   c = cols[e];
    const float v = vals[e];

    const float2 p = *(const float2*)(prod + (size_t)c * GCN_OUT_DIM + 2 * lane);
    float m0 = v * p.x;
    float m1 = v * p.y;

    float* o = out + (size_t)r * GCN_OUT_DIM + 2 * lane;
    // Guaranteed hardware fp32 atomic adds (L2-resident destination).
    asm volatile("global_atomic_add_f32 %0, %1, off" : : "v"(o),     "v"(m0) : "memory");
    asm volatile("global_atomic_add_f32 %0, %1, off" : : "v"(o + 1), "v"(m1) : "memory");
}

// ---------------------------------------------------------------------------
// Kernel 4: in-place ReLU on d_out.
// ---------------------------------------------------------------------------
__global__ __launch_bounds__(256) void gcn_relu(float4* __restrict__ out) {
    const unsigned i = blockIdx.x * 256u + threadIdx.x;
    float4 t = out[i];
    t.x = fmaxf(t.x, 0.f);
    t.y = fmaxf(t.y, 0.f);
    t.z = fmaxf(t.z, 0.f);
    t.w = fmaxf(t.w, 0.f);
    out[i] = t;
}

extern "C" void kernel_launch(void* const* d_in, const int* in_sizes, int n_in,
                              void* d_out, int out_size, void* d_ws, size_t ws_size,
                              hipStream_t stream) {
    const float* x  = (const float*)d_in[0];   // [N, 256]
    const int*   er = (const int*)  d_in[1];   // [E]
    const int*   ec = (const int*)  d_in[2];   // [E]
    const float* ev = (const float*)d_in[3];   // [E]
    const float* w  = (const float*)d_in[4];   // [256, 64]

    float* out  = (float*)d_out;               // [N, 64]
    float* prod = (float*)d_ws;                // [N, 64] scratch (25.6 MB)

    // 1) prod = X @ W   (128 rows per block)
    gcn_gemm_xw<<<(GCN_N_NODES + 127) / 128, 256, 0, stream>>>(x, w, prod);

    // 2) out = 0        (6,400,000 floats = 1,600,000 float4s)
    gcn_zero<<<(GCN_N_NODES * GCN_OUT_DIM / 4) / 256, 256, 0, stream>>>((float4*)out);

    // 3) atomic scatter over edges: 32 lanes/edge
    gcn_spmm<<<(unsigned)(((unsigned long long)GCN_N_EDGES * 32ull) / 256ull), 256, 0, stream>>>(
        er, ec, ev, prod, out);

    // 4) relu in place
    gcn_relu<<<(GCN_N_NODES * GCN_OUT_DIM / 4) / 256, 256, 0, stream>>>((float4*)out);
}